// Reg_model_52913997086838
// MI455X (gfx1250) — compile-verified
//
#include <hip/hip_runtime.h>
#include <hip/hip_bf16.h>

// Problem constants (from reference): B=4, C=1, D=128, H=192, W=192
#define BB 4
#define DD 128
#define HH 192
#define WW 192
#define HW (HH * WW)            // 36864
#define DHW (DD * HH * WW)      // 4718592
#define N_2D (BB * HH * WW)     // 147456
#define N_WARP (BB * DHW)       // 18874368
#define OUT_X2D 0
#define OUT_WARP (N_2D)                   // 147456
#define OUT_Y (N_2D + N_WARP)             // 19021824

typedef __attribute__((ext_vector_type(2))) float v2f;
typedef __attribute__((ext_vector_type(8))) float v8f;
typedef __attribute__((ext_vector_type(4))) float f4;

// ---------------------------------------------------------------------------
// Kernel 1: per-batch affine matrix  mat = T2 @ (rigid @ T1)
//   rotation R = Rz(rot_0) @ Ry(rot_1) @ Rx(rot_2)
//   translation = c - R@c + [0, trans_1*H, trans_2*W]
// 12 floats per batch into d_ws: {r00,r01,r02,T0, r10,r11,r12,T1, r20,r21,r22,T2}
// ---------------------------------------------------------------------------
__global__ void build_mats_kernel(const float* __restrict__ rot0,
                                  const float* __restrict__ rot1,
                                  const float* __restrict__ rot2,
                                  const float* __restrict__ tr1,
                                  const float* __restrict__ tr2,
                                  const float* __restrict__ cp,
                                  float* __restrict__ mats) {
    int b = threadIdx.x;
    if (b >= BB) return;
    float az = rot0[b], ay = rot1[b], ax = rot2[b];
    float cz = cosf(az), sz = sinf(az);
    float cy = cosf(ay), sy = sinf(ay);
    float cx = cosf(ax), sx = sinf(ax);
    // R = Rz @ Ry @ Rx
    float r00 = cz * cy;
    float r01 = -sz * cx + cz * sy * sx;
    float r02 =  sz * sx + cz * sy * cx;
    float r10 =  sz * cy;
    float r11 =  cz * cx + sz * sy * sx;
    float r12 = -cz * sx + sz * sy * cx;
    float r20 = -sy;
    float r21 =  cy * sx;
    float r22 =  cy * cx;
    float c0 = cp[b * 3 + 0], c1 = cp[b * 3 + 1], c2 = cp[b * 3 + 2];
    float t1v = tr1[b] * (float)HH;
    float t2v = tr2[b] * (float)WW;
    float T0 = c0 - (r00 * c0 + r01 * c1 + r02 * c2);
    float T1 = c1 - (r10 * c0 + r11 * c1 + r12 * c2) + t1v;
    float T2 = c2 - (r20 * c0 + r21 * c1 + r22 * c2) + t2v;
    float* m = mats + b * 12;
    m[0] = r00; m[1] = r01; m[2]  = r02; m[3]  = T0;
    m[4] = r10; m[5] = r11; m[6]  = r12; m[7]  = T1;
    m[8] = r20; m[9] = r21; m[10] = r22; m[11] = T2;
}

// ---------------------------------------------------------------------------
// Trilinear gather, zero outside, clamped indexing (matches reference).
// The two x-corners are adjacent in memory: fetch each (z,y) row pair with a
// single 8-byte load from a clamped base (4 loads/sample instead of 8).
// Zero-weight corners tolerate any finite value, so only the two boundary
// clamp cases (ix==-1, ix==W-1) need a select.
// ---------------------------------------------------------------------------
__device__ __forceinline__ float trilerp(const float* __restrict__ xb,
                                         float gz, float gy, float gx) {
    float fz = floorf(gz), fy = floorf(gy), fx = floorf(gx);
    float rz = gz - fz, ry = gy - fy, rx = gx - fx;
    int iz = (int)fz, iy = (int)fy, ix = (int)fx;

    const int bxc = ix < 0 ? 0 : (ix > WW - 2 ? WW - 2 : ix);
    const float wx0 = (1.0f - rx) * ((ix >= 0 && ix < WW) ? 1.0f : 0.0f);
    const float wx1 = rx * ((ix + 1 >= 0 && ix + 1 < WW) ? 1.0f : 0.0f);
    const bool sel_hi = (ix > bxc);   // ix==W-1: low corner is clamped x[W-1] = .y
    const bool sel_lo = (ix < bxc);   // ix==-1 : high corner is clamped x[0]  = .x

    float acc = 0.0f;
#pragma unroll
    for (int dz = 0; dz < 2; ++dz) {
        int z = iz + dz;
        float wz = dz ? rz : 1.0f - rz;
        float vz = (z >= 0 && z < DD) ? 1.0f : 0.0f;
        int zc = z < 0 ? 0 : (z > DD - 1 ? DD - 1 : z);
        const float* pz = xb + (size_t)zc * HW;
#pragma unroll
        for (int dy = 0; dy < 2; ++dy) {
            int yv = iy + dy;
            float wy = dy ? ry : 1.0f - ry;
            float vy = (yv >= 0 && yv < HH) ? 1.0f : 0.0f;
            int yc = yv < 0 ? 0 : (yv > HH - 1 ? HH - 1 : yv);
            const float* py = pz + yc * WW + bxc;

            v2f f;
            __builtin_memcpy(&f, py, 2 * sizeof(float));   // b64 gather
            float c0 = sel_hi ? f.y : f.x;
            float c1 = sel_lo ? f.x : f.y;
            float wzy = (wz * wy) * (vz * vy);
            acc += wzy * (c0 * wx0 + c1 * wx1);
        }
    }
    return acc;
}

// ---------------------------------------------------------------------------
// Kernel 2: warp + WMMA depth reduction.
// One wave32 owns (b, h, 16 consecutive w, all 128 d).
// Depth chunks of 4 are fed as the B operand of V_WMMA_F32_16X16X4_F32 with an
// all-ones A; C row 0 accumulates sum_d x_warp[d,h,w] (ones-A makes the result
// invariant to the K<->(vgpr,half) mapping; only N=lane&15 matters).
// ---------------------------------------------------------------------------
__global__ __launch_bounds__(256) void warp_reduce_kernel(
        const float* __restrict__ x,
        const float* __restrict__ mats,
        float* __restrict__ out) {
    const int lane  = threadIdx.x & 31;
    const int wv    = blockIdx.x * (blockDim.x >> 5) + (threadIdx.x >> 5);
    const int wt    = wv % (WW / 16);
    const int h     = (wv / (WW / 16)) % HH;
    const int b     = wv / ((WW / 16) * HH);
    const int n     = lane & 15;
    const int khalf = lane >> 4;          // 0: lanes 0-15, 1: lanes 16-31
    const int w     = wt * 16 + n;

    const float* m = mats + b * 12;
    const float m00 = m[0], m01 = m[1], m02 = m[2],  t0 = m[3];
    const float m10 = m[4], m11 = m[5], m12 = m[6],  t1 = m[7];
    const float m20 = m[8], m21 = m[9], m22 = m[10], t2 = m[11];

    const float fh = (float)h, fw = (float)w;
    const float bz = m01 * fh + m02 * fw + t0;
    const float by = m11 * fh + m12 * fw + t1;
    const float bx = m21 * fh + m22 * fw + t2;

    const float* xb = x + (size_t)b * DHW;
    float* warp_out = out + OUT_WARP + (size_t)b * DHW + (size_t)h * WW + w;

    v8f c = {};
    v2f a = {1.0f, 1.0f};

    for (int d0 = 0; d0 < DD; d0 += 4) {
        // Pull the next depth rows of x toward the WGP (rotation is small, so
        // the gather footprint tracks the identity coordinate).
        int pd = (d0 + 8 < DD) ? d0 + 8 : DD - 1;
        __builtin_prefetch(xb + (size_t)pd * HW + (size_t)h * WW + w, 0, 1);

        const int da = d0 + (khalf << 1);     // this lane's two depth slices
        const float fd = (float)da;
        float gz0 = m00 * fd + bz;
        float gy0 = m10 * fd + by;
        float gx0 = m20 * fd + bx;
        float v0 = trilerp(xb, gz0, gy0, gx0);
        float v1 = trilerp(xb, gz0 + m00, gy0 + m10, gx0 + m20);

        warp_out[(size_t)da * HW]       = v0;
        warp_out[(size_t)(da + 1) * HW] = v1;

        v2f bm = {v0, v1};
        // 8 args: (neg_a, A, neg_b, B, c_mod, C, reuse_a, reuse_b)
        c = __builtin_amdgcn_wmma_f32_16x16x4_f32(
                false, a, false, bm, (short)0, c, false, false);
    }

    // C-matrix VGPR0: lanes 0-15 hold (M=0, N=lane) = sum over all 128 d.
    if (lane < 16) {
        out[OUT_X2D + (size_t)b * HW + (size_t)h * WW + w] =
            c[0] * (1.0f / (float)DD);
    }
}

// ---------------------------------------------------------------------------
// Kernel 3: y passthrough (vectorized copy)
// ---------------------------------------------------------------------------
__global__ __launch_bounds__(256) void copy_y_kernel(const float* __restrict__ y,
                                                     float* __restrict__ out) {
    int i = blockIdx.x * blockDim.x + threadIdx.x;   // in float4 units
    if (i < N_2D / 4) {
        ((f4*)(out + OUT_Y))[i] = ((const f4*)y)[i];
    }
}

extern "C" void kernel_launch(void* const* d_in, const int* in_sizes, int n_in,
                              void* d_out, int out_size, void* d_ws, size_t ws_size,
                              hipStream_t stream) {
    const float* x    = (const float*)d_in[0];
    const float* y    = (const float*)d_in[1];
    const float* rot0 = (const float*)d_in[2];
    const float* rot1 = (const float*)d_in[3];
    const float* rot2 = (const float*)d_in[4];
    const float* tr1  = (const float*)d_in[5];
    const float* tr2  = (const float*)d_in[6];
    const float* cp   = (const float*)d_in[7];
    float* out  = (float*)d_out;
    float* mats = (float*)d_ws;   // 4 * 12 floats

    build_mats_kernel<<<1, 32, 0, stream>>>(rot0, rot1, rot2, tr1, tr2, cp, mats);

    // waves = B * H * (W/16) = 4*192*12 = 9216 ; 8 waves/block -> 1152 blocks
    const int total_waves = BB * HH * (WW / 16);
    const int blocks = total_waves / 8;
    warp_reduce_kernel<<<blocks, 256, 0, stream>>>(x, mats, out);

    copy_y_kernel<<<(N_2D / 4 + 255) / 256, 256, 0, stream>>>(y, out);
}